// PullbackIsotropyLoss_45226005627445
// MI455X (gfx1250) — compile-verified
//
#include <hip/hip_runtime.h>
#include <math.h>
#include <stdint.h>

#define B_DIM 256
#define D_LAT 32
#define HID   512
#define NG    4000
#define KC    32            // W2 K-rows staged per async chunk
#define ALPHA_C 1.0f
#define EPS_C   1e-6f

typedef float v2f __attribute__((ext_vector_type(2)));
typedef float v8f __attribute__((ext_vector_type(8)));

// D = A(16x4 f32) * B(4x16 f32) + C(16x16 f32), full-precision CDNA5 WMMA.
__device__ __forceinline__ v8f wmma4(v2f a, v2f b, v8f c) {
  return __builtin_amdgcn_wmma_f32_16x16x4_f32(false, a, false, b, (short)0, c, false, false);
}

// Pair-packed, bank-swizzled LDS layout for a [K][32] f32 matrix so that the
// two K-consecutive values of a WMMA B-fragment sit in one 8-byte ds_load_b64,
// and the two half-waves (K pairs p and p+1) hit disjoint bank sets.
__device__ __forceinline__ int pk_idx(int k, int i) {
  int p = k >> 1;
  return (p << 6) + (((i << 1) ^ ((p & 1) << 5)) + (k & 1));
}

// Async-stage KC rows x 16 cols of W2 (cols g0..g0+15, rows kk..kk+KC-1) into a
// wave-private LDS buffer using the CDNA5 async global->LDS engine (ASYNCcnt).
// Each b128 op moves 8 rows (32 lanes x 16B); 4 ops per KC=32 chunk.
__device__ __forceinline__ void stage_w2_chunk(const float* __restrict__ W2,
                                               int g0, int kk, int lane,
                                               float* buf) {
  const int row = lane >> 2;   // 0..7
  const int q   = lane & 3;    // 16-byte quarter of a 64B row
  const float* gp = W2 + (size_t)(kk + row) * NG + g0 + (q << 2);
  unsigned lds = (unsigned)(uintptr_t)buf + (unsigned)(row * 64 + q * 16);
#pragma unroll
  for (int i = 0; i < 4; ++i) {
    asm volatile("global_load_async_to_lds_b128 %0, %1, off"
                 :: "v"(lds), "v"(gp) : "memory");
    gp  += (size_t)8 * NG;
    lds += 512;
  }
}
#define WAIT_ASYNC_LE4() asm volatile("s_wait_asynccnt 0x4" ::: "memory")
#define WAIT_ASYNC_0()   asm volatile("s_wait_asynccnt 0x0" ::: "memory")

// ---------------- Kernel 1: h = relu(z @ W1 + b1) ----------------
__global__ void __launch_bounds__(256) h_kernel(const float* __restrict__ z,
                                                const float* __restrict__ W1,
                                                const float* __restrict__ b1,
                                                float* __restrict__ h) {
  const int b = blockIdx.x;
  const int t = threadIdx.x;
  __shared__ float zs[D_LAT];
  if (t < D_LAT) zs[t] = z[b * D_LAT + t];
  __syncthreads();
  for (int k = t; k < HID; k += 256) {
    float acc = b1[k];
#pragma unroll
    for (int i = 0; i < D_LAT; ++i) acc += zs[i] * W1[i * HID + k];
    h[(size_t)b * HID + k] = fmaxf(acc, 0.0f);
  }
}

// ---------------- Kernel 2: pre2 GEMM (WMMA) + sqrt(c) ----------------
// One wave per 16x16 (batch x gene) tile; 500 blocks * 8 waves = 4000 tiles.
__global__ void __launch_bounds__(256) sc_kernel(const float* __restrict__ h,
                                                 const float* __restrict__ W2,
                                                 const float* __restrict__ b2,
                                                 const float* __restrict__ log_theta,
                                                 float* __restrict__ sc) {
  const int t    = threadIdx.x;
  const int wid  = blockIdx.x * 8 + (t >> 5);
  const int lane = t & 31;
  const int L    = lane & 15;
  const int hi   = lane >> 4;
  const int klo  = hi << 1;            // lanes 0-15 -> K=kk..kk+1, lanes 16-31 -> K=kk+2..kk+3
  const int b0   = (wid & 15) << 4;
  const int g0   = (wid >> 4) << 4;

  v8f acc = {};
  const float* hp  = h  + (size_t)(b0 + L) * HID + klo;
  const float* w2p = W2 + (size_t)klo * NG + (g0 + L);
#pragma unroll 4
  for (int kk = 0; kk < HID; kk += 4) {
    float2 av = *(const float2*)(hp + kk);
    v2f a; a.x = av.x; a.y = av.y;
    v2f bb; bb.x = w2p[(size_t)kk * NG]; bb.y = w2p[(size_t)(kk + 1) * NG];
    acc = wmma4(a, bb, acc);
  }
  const int g = g0 + L;
  const float bias = b2[g];
  const float th   = __expf(log_theta[g]);
#pragma unroll
  for (int r = 0; r < 8; ++r) {
    const int bb = b0 + r + (hi << 3);
    float pre = acc[r] + bias;
    float mu  = (pre > 20.0f) ? pre : log1pf(__expf(pre));  // softplus
    float s   = 1.0f / (1.0f + __expf(-pre));               // sigmoid
    float wgt = th / (mu * (mu + th) + EPS_C);
    sc[(size_t)bb * NG + g] = s * sqrtf(wgt);               // sqrt(c)
  }
}

// ---------------- Kernel 3: per-batch Gram via WMMA, loss_b ----------------
__global__ void __launch_bounds__(256) gram_kernel(const float* __restrict__ W1,
                                                   const float* __restrict__ W2,
                                                   const float* __restrict__ h,
                                                   const float* __restrict__ sc,
                                                   float* __restrict__ losses) {
  __shared__ float Wm[HID * D_LAT];        // 64KB masked W1^T (pair-packed); reused for reduction
  __shared__ float Ubuf[8][16 * D_LAT];    // 16KB per-wave sqrt(c)*M tile
  __shared__ float W2s[8][2][KC * 16];     // 32KB per-wave double-buffered async W2 staging

  const int b    = blockIdx.x;
  const int t    = threadIdx.x;
  const int w    = t >> 5;
  const int lane = t & 31;
  const int L    = lane & 15;
  const int hi   = lane >> 4;
  const int klo  = hi << 1;

  const float* hrow = h + (size_t)b * HID;
  // Stage Wm[k][i] = (h[b,k] > 0) ? W1[i][k] : 0   (relu' mask applied once)
  for (int idx = t; idx < HID * D_LAT; idx += 256) {
    int k = idx & (HID - 1);
    int i = idx >> 9;
    float v = (hrow[k] > 0.0f) ? W1[i * HID + k] : 0.0f;
    Wm[pk_idx(k, i)] = v;
  }
  __syncthreads();

  v8f g00 = {}, g01 = {}, g10 = {}, g11 = {};
  float* U = &Ubuf[w][0];
  float* buf0 = &W2s[w][0][0];
  float* buf1 = &W2s[w][1][0];
  const float* scrow = sc + (size_t)b * NG;

  for (int tt = 0; tt < 32; ++tt) {
    const int g0 = (w + 8 * tt) * 16;   // wave-uniform -> WMMA sees full EXEC
    if (g0 >= NG) break;

    // M_tile[16 genes x 32 lat] = W2_tile^T @ Wm, A-operand streamed through
    // the async DMA engine with double buffering.
    v8f m0 = {}, m1 = {};
    stage_w2_chunk(W2, g0, 0, lane, buf0);
    for (int c = 0; c < HID / KC; ++c) {
      if (c + 1 < HID / KC) {
        stage_w2_chunk(W2, g0, (c + 1) * KC, lane, (c & 1) ? buf0 : buf1);
        WAIT_ASYNC_LE4();               // chunk c fully landed in LDS
      } else {
        WAIT_ASYNC_0();
      }
      const float* bc = (c & 1) ? buf1 : buf0;
#pragma unroll
      for (int kl = 0; kl < KC; kl += 4) {
        const int k = c * KC + kl;      // global K row of this step
        v2f a;
        a.x = bc[(kl + klo) * 16 + L];
        a.y = bc[(kl + klo + 1) * 16 + L];
        const int p   = (k + klo) >> 1;
        const int low = (L << 1) ^ ((p & 1) << 5);
        float2 bv0 = *(const float2*)&Wm[(p << 6) + low];          // i = 0..15
        float2 bv1 = *(const float2*)&Wm[(p << 6) + (low ^ 32)];   // i = 16..31
        v2f bA; bA.x = bv0.x; bA.y = bv0.y;
        v2f bB; bB.x = bv1.x; bB.y = bv1.y;
        m0 = wmma4(a, bA, m0);
        m1 = wmma4(a, bB, m1);
      }
    }

    // U = sqrt(c) * M, stashed in per-wave LDS (same packed layout over gene dim)
#pragma unroll
    for (int r = 0; r < 8; ++r) {
      const int m = r + (hi << 3);          // gene-local row
      const float s = scrow[g0 + m];
      U[pk_idx(m, L)]      = m0[r] * s;
      U[pk_idx(m, L + 16)] = m1[r] * s;
    }

    // Gram += U^T U : K=16 genes in 4 WMMA steps; A-frag(I) == B-frag(J=I)
#pragma unroll
    for (int kk = 0; kk < 16; kk += 4) {
      const int p   = (kk + klo) >> 1;
      const int low = (L << 1) ^ ((p & 1) << 5);
      float2 f0v = *(const float2*)&U[(p << 6) + low];
      float2 f1v = *(const float2*)&U[(p << 6) + (low ^ 32)];
      v2f f0; f0.x = f0v.x; f0.y = f0v.y;
      v2f f1; f1.x = f1v.x; f1.y = f1v.y;
      g00 = wmma4(f0, f0, g00);
      g01 = wmma4(f0, f1, g01);
      g10 = wmma4(f1, f0, g10);
      g11 = wmma4(f1, f1, g11);
    }
  }
  __syncthreads();   // all waves done with Wm -> reuse as reduction buffer

  float* red = Wm;   // 8 waves * 1024 partial Gram entries = 8192 floats
#pragma unroll
  for (int r = 0; r < 8; ++r) {
    const int i0 = r + (hi << 3);
    red[w * 1024 + (i0)      * 32 + L]      = g00[r];
    red[w * 1024 + (i0)      * 32 + L + 16] = g01[r];
    red[w * 1024 + (i0 + 16) * 32 + L]      = g10[r];
    red[w * 1024 + (i0 + 16) * 32 + L + 16] = g11[r];
  }
  __syncthreads();

  float sq = 0.0f;
  for (int e = t; e < 1024; e += 256) {
    float g = 0.0f;
#pragma unroll
    for (int ww = 0; ww < 8; ++ww) g += red[ww * 1024 + e];
    const int i = e >> 5, j = e & 31;
    const float d = g - ((i == j) ? ALPHA_C : 0.0f);
    sq += d * d;
  }
  __syncthreads();
  red[t] = sq;
  __syncthreads();
  for (int st = 128; st > 0; st >>= 1) {
    if (t < st) red[t] += red[t + st];
    __syncthreads();
  }
  if (t == 0) losses[b] = red[0];
}

// ---------------- Kernel 4: mean over batches ----------------
__global__ void __launch_bounds__(256) loss_kernel(const float* __restrict__ losses,
                                                   float* __restrict__ out) {
  __shared__ float s[256];
  const int t = threadIdx.x;
  s[t] = losses[t];
  __syncthreads();
  for (int st = 128; st > 0; st >>= 1) {
    if (t < st) s[t] += s[t + st];
    __syncthreads();
  }
  if (t == 0) out[0] = s[0] * (1.0f / (float)B_DIM);
}

extern "C" void kernel_launch(void* const* d_in, const int* in_sizes, int n_in,
                              void* d_out, int out_size, void* d_ws, size_t ws_size,
                              hipStream_t stream) {
  const float* z         = (const float*)d_in[0];
  const float* W1        = (const float*)d_in[1];
  const float* b1        = (const float*)d_in[2];
  const float* W2        = (const float*)d_in[3];
  const float* b2        = (const float*)d_in[4];
  const float* log_theta = (const float*)d_in[5];
  float* out = (float*)d_out;

  float* h      = (float*)d_ws;                 // 256*512
  float* sc     = h + (size_t)B_DIM * HID;      // 256*4000
  float* losses = sc + (size_t)B_DIM * NG;      // 256

  h_kernel   <<<B_DIM, 256, 0, stream>>>(z, W1, b1, h);
  sc_kernel  <<<500,   256, 0, stream>>>(h, W2, b2, log_theta, sc);
  gram_kernel<<<B_DIM, 256, 0, stream>>>(W1, W2, h, sc, losses);
  loss_kernel<<<1,     256, 0, stream>>>(losses, out);
}